// FSMRE_28114855920237
// MI455X (gfx1250) — compile-verified
//
#include <hip/hip_runtime.h>

typedef __attribute__((ext_vector_type(2))) float v2f;
typedef __attribute__((ext_vector_type(8))) float v8f;

#define S_ 32
#define E_ 48
#define H_ 512
#define L_ 16
#define M_ (S_*E_)          // 1536 rows in the flattened GEMM
#define EE_ (E_*E_)         // 2304

// ---------------------------------------------------------------------------
// Kernel 0: c[l] = ic[l]/(total - ic[l]) - p2[l]
// (p2[l] = sum over 2H of prototype[l]^2; n2 terms cancel in the softmax)
// One block of 256 threads: 16 threads per l, float4 partial sums + shfl reduce.
// ---------------------------------------------------------------------------
__global__ __launch_bounds__(256)
void FSMRE_prep_kernel(const float* __restrict__ proto,
                       const float* __restrict__ ic,
                       float* __restrict__ c) {
    const int t    = threadIdx.x;
    const int l    = t >> 4;      // 0..15
    const int part = t & 15;      // 0..15
    // prototype row = 1024 floats = 256 float4; 16 float4 per partial
    const float4* p4 = (const float4*)(proto + l * (2 * H_)) + part * 16;
    float p2 = 0.f;
    #pragma unroll 4
    for (int i = 0; i < 16; ++i) {
        float4 v = p4[i];
        p2 += v.x * v.x + v.y * v.y + v.z * v.z + v.w * v.w;
    }
    // reduce across the 16-lane group (xor masks < 16 stay inside the group)
    for (int off = 8; off; off >>= 1) p2 += __shfl_xor(p2, off, 32);
    if (part == 0) {
        float total = 0.f;
        for (int j = 0; j < L_; ++j) total += ic[j];
        float icl = ic[l];
        c[l] = icl / (total - icl) - p2;
    }
}

// ---------------------------------------------------------------------------
// Kernel 1: a = emb(1536x512) * p_head^T, b = emb * p_tail^T via
// V_WMMA_F32_16X16X4_F32 (f32 inputs, f32 accumulate -> exact vs reference).
// One wave (32 threads) per 16-row tile; 96 tiles. Per K=4 step:
//   A frag: lane holds row (lane&15), float2 at col k0 + 2*(lane>>4)
//   B frag: lane holds col l = (lane&15), B[k][n] = proto[n][k] -> same pattern
// ---------------------------------------------------------------------------
__global__ __launch_bounds__(32)
void FSMRE_gemm_kernel(const float* __restrict__ emb,
                       const float* __restrict__ proto,
                       float* __restrict__ wa,
                       float* __restrict__ wb) {
    const int lane = threadIdx.x;      // 0..31 (wave32)
    const int half = lane >> 4;        // 0 or 1
    const int rl   = lane & 15;        // A: row-in-tile, B: column (= l)
    const int tile = blockIdx.x;       // 0..95

    const float* aptr = emb   + (tile * 16 + rl) * H_ + 2 * half;
    const float* hptr = proto + rl * (2 * H_)        + 2 * half;  // p_head row
    const float* tptr = hptr + H_;                                 // p_tail row

    v8f accA = {};
    v8f accB = {};
    #pragma unroll 8
    for (int k0 = 0; k0 < H_; k0 += 4) {
        v2f av = *(const v2f*)(aptr + k0);
        v2f bh = *(const v2f*)(hptr + k0);
        v2f bt = *(const v2f*)(tptr + k0);
        // 8 args: (neg_a, A, neg_b, B, c_mod, C, reuse_a, reuse_b)
        accA = __builtin_amdgcn_wmma_f32_16x16x4_f32(
            false, av, false, bh, (short)0, accA, false, false);
        accB = __builtin_amdgcn_wmma_f32_16x16x4_f32(
            false, av, false, bt, (short)0, accB, false, false);
    }

    // C/D layout: VGPR r -> M = r + 8*(lane>=16), N = lane&15
    #pragma unroll
    for (int r = 0; r < 8; ++r) {
        int m = tile * 16 + r + 8 * half;
        wa[m * L_ + rl] = accA[r];
        wb[m * L_ + rl] = accB[r];
    }
}

// ---------------------------------------------------------------------------
// Kernel 2 (the bandwidth kernel, ~75.5 MB of stores):
// each block handles 16 consecutive triples g=(s,e1,e2); thread t computes one
// logit (triple = t>>4, l = t&15), softmax over 16-lane shfl groups, diagonal
// zeroing, then the whole block streams a contiguous 16 KB chunk as float4.
// ---------------------------------------------------------------------------
__global__ __launch_bounds__(256)
void FSMRE_softmax_bcast_kernel(const float* __restrict__ wa,
                                const float* __restrict__ wb,
                                const float* __restrict__ c,
                                float* __restrict__ out) {
    __shared__ float lds[256];
    const int t  = threadIdx.x;
    const int g  = blockIdx.x * 16 + (t >> 4);   // triple index
    const int l  = t & 15;
    const int s  = g / EE_;
    const int r  = g % EE_;
    const int e1 = r / E_;
    const int e2 = r % E_;
    const int m1 = s * E_ + e1;
    const int m2 = s * E_ + e2;

    float logit = 2.f * (wa[m1 * L_ + l] + wb[m2 * L_ + l]) + c[l];

    float mx = logit;
    for (int off = 8; off; off >>= 1) mx = fmaxf(mx, __shfl_xor(mx, off, 32));
    float ex = __expf(logit - mx);
    float sum = ex;
    for (int off = 8; off; off >>= 1) sum += __shfl_xor(sum, off, 32);

    float pred = (e1 == e2) ? 0.f : ex / sum;
    lds[t] = pred;
    __syncthreads();

    // 16 triples * 256 floats = 4096 floats = 1024 float4 per block, contiguous.
    // out element (triple, e) = lds[triple*16 + (e & 15)]  (l1-broadcast)
    const float4* l4 = (const float4*)lds;
    float4* o4 = (float4*)out + (size_t)blockIdx.x * 1024;
    #pragma unroll
    for (int i = 0; i < 4; ++i) {
        int idx = t + i * 256;                       // float4 index in chunk
        o4[idx] = l4[((idx >> 6) << 2) + (idx & 3)]; // triple*4 + quad
    }
}

// ---------------------------------------------------------------------------
extern "C" void kernel_launch(void* const* d_in, const int* in_sizes, int n_in,
                              void* d_out, int out_size, void* d_ws, size_t ws_size,
                              hipStream_t stream) {
    const float* emb   = (const float*)d_in[0];   // (S,E,H)  f32
    const float* proto = (const float*)d_in[1];   // (L,2H)   f32
    const float* ic    = (const float*)d_in[2];   // (L,)     f32

    float* wa  = (float*)d_ws;          // 1536*16 floats
    float* wb  = wa + M_ * L_;          // 1536*16 floats
    float* c   = wb + M_ * L_;          // 16 floats
    float* out = (float*)d_out;

    FSMRE_prep_kernel<<<1, 256, 0, stream>>>(proto, ic, c);
    FSMRE_gemm_kernel<<<M_ / 16, 32, 0, stream>>>(emb, proto, wa, wb);
    FSMRE_softmax_bcast_kernel<<<(S_ * EE_) / 16, 256, 0, stream>>>(wa, wb, c, out);
}